// ProposalLayer1_45397804319432
// MI455X (gfx1250) — compile-verified
//
#include <hip/hip_runtime.h>
#include <hip/hip_bf16.h>
#include <stdint.h>

// ---------------- problem constants (match setup_inputs) ----------------
#define B_SZ    4
#define N_PTS   16384      // points per scene (power of 2 -> bitonic)
#define REG_C   76
#define PRE_N   9000
#define POST_N  512
#define NMS_TH  0.85f

#define SORT_THREADS 1024
#define NMS_THREADS  1024
#define FULL_WORDS   281            // words 0..280 fully inside [0,9000)
#define SUP_WORDS    288            // ceil(9000/32)=282, padded

// ---------------- CDNA5 async global->LDS (ASYNCcnt path) ----------------
// gfx1250: GLOBAL_LOAD_ASYNC_TO_LDS_B128 moves 16B/lane from global memory
// into LDS without touching VGPR data; tracked with ASYNCcnt. Generic
// pointers to LDS carry the wave-relative LDS offset in addr[31:0], which is
// exactly what the VDST VGPR of the async op expects.
__device__ __forceinline__ void async_g2l_b128(void* lds_dst, const void* g_src) {
  uint32_t loff = (uint32_t)(uintptr_t)lds_dst;
  unsigned long long ga = (unsigned long long)(uintptr_t)g_src;
  asm volatile("global_load_async_to_lds_b128 %0, %1, off"
               :: "v"(loff), "v"(ga) : "memory");
}
__device__ __forceinline__ void async_wait0() {
  asm volatile("s_wait_asynccnt 0" ::: "memory");
}

#if defined(__has_builtin)
#if __has_builtin(__builtin_amdgcn_ballot_w32)
#define BALLOT32(p) __builtin_amdgcn_ballot_w32(p)
#endif
#endif
#ifndef BALLOT32
#define BALLOT32(p) ((unsigned)__ballot(p))
#endif

// =====================================================================
// Kernel 1: per-scene full bitonic sort of (score desc, idx asc) in LDS.
// One 1024-thread workgroup per scene; 16384 keys + 16384 idx = 128KB LDS
// (dynamic). Emits sorted top PRE_N scores + source indices.
// =====================================================================
__global__ __launch_bounds__(SORT_THREADS)
void pl_sort_kernel(const float* __restrict__ scores,
                    float* __restrict__ out_s,
                    int* __restrict__ out_i) {
  extern __shared__ __align__(16) char smem[];
  float* skey = (float*)smem;                       // N_PTS floats
  int*   sidx = (int*)(smem + N_PTS * sizeof(float));
  const int b   = blockIdx.x;
  const int tid = threadIdx.x;
  const float* srow = scores + (size_t)b * N_PTS;

  // async DMA the score row straight into the LDS key array (16B granules)
  for (int c = tid; c < N_PTS / 4; c += SORT_THREADS)
    async_g2l_b128(&skey[4 * c], &srow[4 * c]);
  for (int e = tid; e < N_PTS; e += SORT_THREADS) sidx[e] = e;
  async_wait0();
  __syncthreads();

  // bitonic sort: final order = descending score, index ascending on ties
  for (unsigned k = 2; k <= (unsigned)N_PTS; k <<= 1) {
    for (unsigned j = k >> 1; j >= 1u; j >>= 1) {
      for (unsigned t = tid; t < (unsigned)(N_PTS / 2); t += SORT_THREADS) {
        unsigned i = ((t & ~(j - 1u)) << 1) | (t & (j - 1u));
        unsigned p = i | j;
        float sa = skey[i], sb = skey[p];
        int   ia = sidx[i], ib = sidx[p];
        bool before = (sa > sb) || (sa == sb && ia < ib);  // a ranks first
        bool up = ((i & k) == 0u);
        if (before != up) {
          skey[i] = sb; skey[p] = sa;
          sidx[i] = ib; sidx[p] = ia;
        }
      }
      __syncthreads();
    }
  }

  for (int r = tid; r < PRE_N; r += SORT_THREADS) {
    out_s[b * PRE_N + r] = skey[r];
    out_i[b * PRE_N + r] = sidx[r];
  }
}

// =====================================================================
// Kernel 2: decode only the top-PRE_N gathered rows (argmax bins + residual
// selection fully unrolled so everything stays in registers, no scratch).
// Writes padded box8 (x,y,z,h,w,l,ry,0) and BEV float4 (x1,y1,x2,y2).
// =====================================================================
__global__ __launch_bounds__(256)
void pl_decode_kernel(const float* __restrict__ reg,
                      const float* __restrict__ xyz,
                      const int* __restrict__ sidx,
                      float* __restrict__ box8,
                      float* __restrict__ bev4) {
  int g = blockIdx.x * blockDim.x + threadIdx.x;
  if (g >= B_SZ * PRE_N) return;
  int b = g / PRE_N;
  int idx = sidx[g];

  const float* rr = reg + ((size_t)b * N_PTS + idx) * REG_C;
  __builtin_prefetch(rr, 0, 3);          // gfx1250: global_prefetch_b8

  float v[REG_C];
  const float4* r4 = (const float4*)rr;  // row stride 304B -> 16B aligned
#pragma unroll
  for (int q = 0; q < REG_C / 4; ++q) {
    float4 t = r4[q];
    v[4 * q + 0] = t.x; v[4 * q + 1] = t.y;
    v[4 * q + 2] = t.z; v[4 * q + 3] = t.w;
  }
  const float* px = xyz + ((size_t)b * N_PTS + idx) * 3;
  float rox = px[0], roy = px[1], roz = px[2];

  // layout: [0:12) xbin [12:24) zbin [24:36) xres [36:48) zres [48] yoff
  //         [49:61) rybin [61:73) ryres [73:76) size
  float bmx = v[0];  int xb = 0; float xres = v[24];
  float bmz = v[12];             float zres = v[36];
  float bmr = v[49]; int rb = 0; float rres = v[61];
  int zb = 0;
#pragma unroll
  for (int k = 1; k < 12; ++k) {
    if (v[k]      > bmx) { bmx = v[k];      xb = k; xres = v[24 + k]; }
    if (v[12 + k] > bmz) { bmz = v[12 + k]; zb = k; zres = v[36 + k]; }
    if (v[49 + k] > bmr) { bmr = v[49 + k]; rb = k; rres = v[61 + k]; }
  }
  const float bs = 0.5f, scope = 3.0f;
  float posx = (float)xb * bs + 0.5f * bs - scope + xres * bs + rox;
  float posz = (float)zb * bs + 0.5f * bs - scope + zres * bs + roz;
  float posy = roy + v[48];

  const float TWO_PI_F = 6.28318530717958647692f;
  const float apc = TWO_PI_F / 12.0f;
  float ry = fmodf((float)rb * apc + rres * (0.5f * apc), TWO_PI_F);
  if (ry < 0.0f) ry += TWO_PI_F;                 // python floor-mod
  if (ry > 3.14159265358979323846f) ry -= TWO_PI_F;

  const float M0 = 1.52563191462f, M1 = 1.62856739989f, M2 = 3.88311640418f;
  float h = v[73] * M0 + M0;
  float w = v[74] * M1 + M1;
  float l = v[75] * M2 + M2;
  posy += 0.5f * h;                              // props[:,1] += h/2

  float* ob = box8 + (size_t)g * 8;
  ob[0] = posx; ob[1] = posy; ob[2] = posz;
  ob[3] = h;    ob[4] = w;    ob[5] = l;
  ob[6] = ry;   ob[7] = 0.0f;

  float4* obv = (float4*)(bev4 + (size_t)g * 4);
  *obv = make_float4(posx - 0.5f * l, posz - 0.5f * w,
                     posx + 0.5f * l, posz + 0.5f * w);
}

// =====================================================================
// Kernel 3: greedy axis-aligned NMS with early exit at POST_N keepers.
// One workgroup per scene. BEV boxes async-DMA'd to LDS (144KB), wave32
// ballot builds the suppression bitmask (no atomics: each wave owns the
// word covering its 32 aligned boxes), thread 0 bit-scans for the next
// unsuppressed box. Then scatters kept boxes/scores and zero-pads d_out.
// =====================================================================
__global__ __launch_bounds__(NMS_THREADS)
void pl_nms_kernel(const float* __restrict__ bev4,
                   const float* __restrict__ box8,
                   const float* __restrict__ ssc,
                   float* __restrict__ out) {
  extern __shared__ __align__(16) char smem[];
  float4*   bev  = (float4*)smem;                          // PRE_N
  unsigned* sup  = (unsigned*)(smem + (size_t)PRE_N * sizeof(float4));
  int*      kept = (int*)(sup + SUP_WORDS);                // POST_N
  int*      ctrl = kept + POST_N;                          // [0]=cur [1]=kcnt

  const int b   = blockIdx.x;
  const int tid = threadIdx.x;

  const float4* src = (const float4*)(bev4 + (size_t)b * PRE_N * 4);
  for (int r = tid; r < PRE_N; r += NMS_THREADS)
    async_g2l_b128(&bev[r], &src[r]);
  for (int wdx = tid; wdx < SUP_WORDS; wdx += NMS_THREADS)
    sup[wdx] = (wdx < FULL_WORDS) ? 0u
             : (wdx == FULL_WORDS ? 0xFFFFFF00u   // boxes 9000..9023 invalid
                                  : 0xFFFFFFFFu);
  if (tid == 0) { ctrl[0] = 0; ctrl[1] = 0; }
  async_wait0();
  __syncthreads();

  while (true) {
    int i  = ctrl[0];
    int kc = ctrl[1];
    if (i >= PRE_N || kc >= POST_N) break;

    float4 bi = bev[i];
    float ai = (bi.z - bi.x) * (bi.w - bi.y);

    for (int base = 0; base < PRE_N; base += NMS_THREADS) {
      int j = base + tid;
      bool pred = false;
      if (j < PRE_N && j > i) {
        float4 bj = bev[j];
        float xx1 = fmaxf(bi.x, bj.x), yy1 = fmaxf(bi.y, bj.y);
        float xx2 = fminf(bi.z, bj.z), yy2 = fminf(bi.w, bj.w);
        float inter = fmaxf(xx2 - xx1, 0.0f) * fmaxf(yy2 - yy1, 0.0f);
        float aj = (bj.z - bj.x) * (bj.w - bj.y);
        float iou = inter / fmaxf(ai + aj - inter, 1e-8f);
        pred = iou > NMS_TH;
      }
      unsigned m = BALLOT32(pred);
      if ((tid & 31) == 0 && m) sup[j >> 5] |= m;   // j is 32-aligned here
    }
    __syncthreads();

    if (tid == 0) {
      kept[kc] = i;
      ctrl[1] = kc + 1;
      int nj = i + 1;                                // find next unsuppressed
      int wdx = nj >> 5;
      unsigned lowmask = (nj & 31) ? ((1u << (nj & 31)) - 1u) : 0u;
      unsigned word = sup[wdx] | lowmask;
      while (word == 0xFFFFFFFFu && wdx < FULL_WORDS) { ++wdx; word = sup[wdx]; }
      ctrl[0] = (word == 0xFFFFFFFFu) ? PRE_N
                                      : ((wdx << 5) + __builtin_ctz(~word));
    }
    __syncthreads();
  }

  // scatter kept boxes / scores; zero-pad the rest (d_out is poisoned)
  int kc = ctrl[1];
  if (kc > POST_N) kc = POST_N;
  const float* boxb = box8 + (size_t)b * PRE_N * 8;
  const float* scb  = ssc  + (size_t)b * PRE_N;
  float* obox = out + (size_t)b * POST_N * 7;
  float* oscr = out + (size_t)B_SZ * POST_N * 7 + (size_t)b * POST_N;
  for (int r = tid; r < POST_N; r += NMS_THREADS) {
    float bx[7] = {0.f, 0.f, 0.f, 0.f, 0.f, 0.f, 0.f};
    float s = 0.f;
    if (r < kc) {
      int ksrc = kept[r];
      const float* sb7 = boxb + (size_t)ksrc * 8;
#pragma unroll
      for (int c = 0; c < 7; ++c) bx[c] = sb7[c];
      s = scb[ksrc];
    }
#pragma unroll
    for (int c = 0; c < 7; ++c) obox[(size_t)r * 7 + c] = bx[c];
    oscr[r] = s;
  }
}

// =====================================================================
// Host launcher
// =====================================================================
extern "C" void kernel_launch(void* const* d_in, const int* in_sizes, int n_in,
                              void* d_out, int out_size, void* d_ws, size_t ws_size,
                              hipStream_t stream) {
  const float* scores = (const float*)d_in[0];   // (4,16384)
  const float* reg    = (const float*)d_in[1];   // (4,16384,76)
  const float* xyz    = (const float*)d_in[2];   // (4,16384,3)
  // d_in[3] (gt_boxes3d) unused by the reference outputs
  float* out = (float*)d_out;                    // boxes(4,512,7) ++ scores(4,512)

  // workspace carve (all 16B aligned): scores | idx | box8 | bev4
  float* ws_scores = (float*)d_ws;                          // B*PRE_N
  int*   ws_idx    = (int*)(ws_scores + B_SZ * PRE_N);      // B*PRE_N
  float* ws_box8   = (float*)(ws_idx + B_SZ * PRE_N);       // B*PRE_N*8
  float* ws_bev4   = ws_box8 + (size_t)B_SZ * PRE_N * 8;    // B*PRE_N*4

  const size_t sort_lds = (size_t)N_PTS * (sizeof(float) + sizeof(int)); // 128KB
  const size_t nms_lds  = (size_t)PRE_N * sizeof(float4)
                        + SUP_WORDS * sizeof(unsigned)
                        + POST_N * sizeof(int) + 4 * sizeof(int);        // ~147KB

  pl_sort_kernel<<<B_SZ, SORT_THREADS, sort_lds, stream>>>(scores, ws_scores, ws_idx);

  int total = B_SZ * PRE_N;
  pl_decode_kernel<<<(total + 255) / 256, 256, 0, stream>>>(reg, xyz, ws_idx,
                                                            ws_box8, ws_bev4);

  pl_nms_kernel<<<B_SZ, NMS_THREADS, nms_lds, stream>>>(ws_bev4, ws_box8,
                                                        ws_scores, out);
}